// RouteGraphEncoder_85298050499127
// MI455X (gfx1250) — compile-verified
//
#include <hip/hip_runtime.h>
#include <hip/hip_bf16.h>
#include <math.h>

typedef __attribute__((ext_vector_type(2))) float v2f;
typedef __attribute__((ext_vector_type(8))) float v8f;
typedef int v4i __attribute__((vector_size(16)));

namespace {
constexpr int B_  = 32;
constexpr int L_  = 2048;
constexpr int D_  = 128;
constexpr int H_  = 4;
constexpr int NL_ = 2;
constexpr int EP_ = 32768;
constexpr int N_  = B_ * L_;    // 65536 nodes
constexpr int E_  = B_ * EP_;   // 1048576 edges
}

__device__ __forceinline__ float lrelu02(float x) { return x > 0.f ? x : 0.2f * x; }

__device__ __forceinline__ void edge_nodes(const int* __restrict__ adj, int e,
                                           int& s, int& d) {
  int b = e >> 15;              // e / EP_
  int i = e & (EP_ - 1);
  s = adj[((b << 1) + 0) * EP_ + i] + b * L_;
  d = adj[((b << 1) + 1) * EP_ + i] + b * L_;
}

// ---------------------------------------------------------------------------
// x = h @ W   (N x 128) * (128 x 128), fp32 WMMA 16x16x4.
//
// Block = 256 threads (8 waves). The full 128x128 W panel (64 KB) is staged
// into LDS once per block — via async global->LDS (ASYNCcnt) when the
// builtin exists — then each wave computes a 16x128 row strip: A fragments
// are loaded from global exactly once and reused across all 8 column tiles
// (8 v8f accumulators), B fragments come from LDS.
//
// A layout (16x4 f32): lanes 0-15 hold row M=lane, vgpr0=K0, vgpr1=K1;
//                      lanes 16-31 hold K2/K3.
// C/D layout: vgpr i -> row (i + 8*half), col = lane%16.
// ---------------------------------------------------------------------------
__global__ void __launch_bounds__(256)
gat_gemm_wmma(const float* __restrict__ hsrc, const float* __restrict__ Wl,
              float* __restrict__ x) {
  __shared__ float sW[D_ * D_];          // 64 KB

  // ---- stage W into LDS: 4096 float4s over 256 threads (16 each) ----
#pragma unroll
  for (int i = 0; i < 16; ++i) {
    int f = threadIdx.x + i * 256;       // float4 index
    const float* g = Wl + f * 4;
    float* l = sW + f * 4;
#if __has_builtin(__builtin_amdgcn_global_load_async_to_lds_b128)
    __builtin_amdgcn_global_load_async_to_lds_b128(
        (__attribute__((address_space(1))) v4i*)(uintptr_t)g,
        (__attribute__((address_space(3))) v4i*)(uintptr_t)l,
        0, 0);
#else
    *(float4*)l = *(const float4*)g;
#endif
  }
#if __has_builtin(__builtin_amdgcn_global_load_async_to_lds_b128)
#if __has_builtin(__builtin_amdgcn_s_wait_asynccnt)
  __builtin_amdgcn_s_wait_asynccnt(0);
#else
  asm volatile("s_wait_asynccnt 0" ::: "memory");
#endif
#endif
  __syncthreads();

  const int wave = threadIdx.x >> 5;
  const int lane = threadIdx.x & 31;
  const int half = lane >> 4;
  const int sub  = lane & 15;
  const int row0 = blockIdx.x * 128 + wave * 16;

  const float* __restrict__ arow = hsrc + (size_t)(row0 + sub) * D_;

  v8f acc[8] = {};
  for (int k = 0; k < D_; k += 4) {
    v2f a = *(const v2f*)(arow + k + 2 * half);         // contiguous K pair
    const float* b0 = sW + (k + 2 * half) * D_ + sub;   // row k (+2 for hi half)
    const float* b1 = b0 + D_;                          // row k+1 (k+3)
#pragma unroll
    for (int j = 0; j < 8; ++j) {
      v2f b;
      b.x = b0[j * 16];
      b.y = b1[j * 16];
      acc[j] = __builtin_amdgcn_wmma_f32_16x16x4_f32(false, a, false, b,
                                                     (short)0, acc[j],
                                                     false, false);
    }
  }

#pragma unroll
  for (int j = 0; j < 8; ++j) {
    float* __restrict__ orow =
        x + (size_t)(row0 + 8 * half) * D_ + j * 16 + sub;
#pragma unroll
    for (int i = 0; i < 8; ++i)
      orow[(size_t)i * D_] = acc[j][i];
  }
}

// ---------------------------------------------------------------------------
// Per-(node,head) attention logits: ls = <x_nh, a_s>, ld = <x_nh, a_d>
// ---------------------------------------------------------------------------
__global__ void gat_logits(const float* __restrict__ x,
                           const float* __restrict__ a_s,
                           const float* __restrict__ a_d,
                           float* __restrict__ ls, float* __restrict__ ld) {
  int i = blockIdx.x * blockDim.x + threadIdx.x;
  if (i >= N_ * H_) return;
  int n = i >> 2, h = i & 3;
  const float* xp  = x + (size_t)n * D_ + h * 32;
  const float* asp = a_s + h * 32;
  const float* adp = a_d + h * 32;
  float s = 0.f, d = 0.f;
#pragma unroll
  for (int c = 0; c < 32; ++c) {
    float v = xp[c];
    s = fmaf(v, asp[c], s);
    d = fmaf(v, adp[c], d);
  }
  ls[i] = s;
  ld[i] = d;
}

// ---------------------------------------------------------------------------
// Per-layer scratch init: segout = 0, m = -inf, s = 0
// ---------------------------------------------------------------------------
__global__ void gat_init(float* __restrict__ segout, float* __restrict__ m,
                         float* __restrict__ ssum) {
  int i = blockIdx.x * blockDim.x + threadIdx.x;
  if (i < N_ * D_) segout[i] = 0.f;
  if (i < N_ * H_) { m[i] = -INFINITY; ssum[i] = 0.f; }
}

// ---------------------------------------------------------------------------
// Segment max over dst: m[dst,h] = max(alpha)  (global_atomic_max_num_f32)
// ---------------------------------------------------------------------------
__global__ void gat_edge_max(const int* __restrict__ adj,
                             const float* __restrict__ ls,
                             const float* __restrict__ ld,
                             float* __restrict__ m) {
  int e = blockIdx.x * blockDim.x + threadIdx.x;
  if (e >= E_) return;
  int s, d;
  edge_nodes(adj, e, s, d);
#pragma unroll
  for (int h = 0; h < H_; ++h) {
    float a = lrelu02(ls[s * H_ + h] + ld[d * H_ + h]);
    atomicMax(&m[d * H_ + h], a);
  }
}

// ---------------------------------------------------------------------------
// Segment sum of exp(alpha - m[dst]) over dst (alpha recomputed; cheaper
// than spilling 16MB of per-edge alphas).
// ---------------------------------------------------------------------------
__global__ void gat_edge_sum(const int* __restrict__ adj,
                             const float* __restrict__ ls,
                             const float* __restrict__ ld,
                             const float* __restrict__ m,
                             float* __restrict__ ssum) {
  int e = blockIdx.x * blockDim.x + threadIdx.x;
  if (e >= E_) return;
  int s, d;
  edge_nodes(adj, e, s, d);
#pragma unroll
  for (int h = 0; h < H_; ++h) {
    float a  = lrelu02(ls[s * H_ + h] + ld[d * H_ + h]);
    float ex = __expf(a - m[d * H_ + h]);
    atomicAdd(&ssum[d * H_ + h], ex);
  }
}

// ---------------------------------------------------------------------------
// Weighted scatter: segout[dst] += w * x[src]. One wave32 per edge; each
// lane owns 4 contiguous channels (one float4 gather + 4 f32 atomics).
// ---------------------------------------------------------------------------
__global__ void gat_edge_scatter(const int* __restrict__ adj,
                                 const float* __restrict__ ls,
                                 const float* __restrict__ ld,
                                 const float* __restrict__ m,
                                 const float* __restrict__ ssum,
                                 const float* __restrict__ x,
                                 float* __restrict__ segout) {
  int lane = threadIdx.x & 31;
  int e = blockIdx.x * (blockDim.x >> 5) + (threadIdx.x >> 5);
  if (e >= E_) return;
  int s, d;
  edge_nodes(adj, e, s, d);
  int c0 = lane << 2;           // channels c0..c0+3, all in one head (32 | 4)
  int h  = c0 >> 5;
  float a = lrelu02(ls[s * H_ + h] + ld[d * H_ + h]);
  float w = __expf(a - m[d * H_ + h]) / (ssum[d * H_ + h] + 1e-16f);
  const float4 xv = *(const float4*)(x + (size_t)s * D_ + c0);
  float* op = segout + (size_t)d * D_ + c0;
  atomicAdd(op + 0, w * xv.x);
  atomicAdd(op + 1, w * xv.y);
  atomicAdd(op + 2, w * xv.z);
  atomicAdd(op + 3, w * xv.w);
}

// ---------------------------------------------------------------------------
// h_out = LayerNorm(h + segout + bias) * gamma + beta. One wave32 per row of
// 128; lane holds 4 values, reductions via wave32 shfl_xor.
// ---------------------------------------------------------------------------
__global__ void gat_ln(const float* __restrict__ hsrc,
                       const float* __restrict__ segout,
                       const float* __restrict__ bias_l,
                       const float* __restrict__ gamma_l,
                       const float* __restrict__ beta_l,
                       float* __restrict__ hout) {
  int lane = threadIdx.x & 31;
  int n = blockIdx.x * (blockDim.x >> 5) + (threadIdx.x >> 5);
  if (n >= N_) return;
  int c0 = lane << 2;
  const float4 hv = *(const float4*)(hsrc + (size_t)n * D_ + c0);
  const float4 sv = *(const float4*)(segout + (size_t)n * D_ + c0);
  const float4 bv = *(const float4*)(bias_l + c0);
  float v0 = hv.x + sv.x + bv.x;
  float v1 = hv.y + sv.y + bv.y;
  float v2 = hv.z + sv.z + bv.z;
  float v3 = hv.w + sv.w + bv.w;
  float sum = v0 + v1 + v2 + v3;
  float sq  = v0 * v0 + v1 * v1 + v2 * v2 + v3 * v3;
#pragma unroll
  for (int off = 16; off > 0; off >>= 1) {
    sum += __shfl_xor(sum, off, 32);
    sq  += __shfl_xor(sq, off, 32);
  }
  float mu  = sum * (1.f / 128.f);
  float var = sq * (1.f / 128.f) - mu * mu;
  float r   = rsqrtf(var + 1e-5f);
  const float4 gv = *(const float4*)(gamma_l + c0);
  const float4 bt = *(const float4*)(beta_l + c0);
  float4 o;
  o.x = (v0 - mu) * r * gv.x + bt.x;
  o.y = (v1 - mu) * r * gv.y + bt.y;
  o.z = (v2 - mu) * r * gv.z + bt.z;
  o.w = (v3 - mu) * r * gv.w + bt.w;
  *(float4*)(hout + (size_t)n * D_ + c0) = o;
}

// ---------------------------------------------------------------------------
extern "C" void kernel_launch(void* const* d_in, const int* in_sizes, int n_in,
                              void* d_out, int out_size, void* d_ws, size_t ws_size,
                              hipStream_t stream) {
  (void)in_sizes; (void)n_in; (void)out_size; (void)ws_size;

  const float* route_emb = (const float*)d_in[0];
  // d_in[1] = route_len (all == L, unused)
  const int*   adj     = (const int*)d_in[2];
  const float* W       = (const float*)d_in[3];
  const float* att_src = (const float*)d_in[4];
  const float* att_dst = (const float*)d_in[5];
  const float* bias    = (const float*)d_in[6];
  const float* gamma   = (const float*)d_in[7];
  const float* beta    = (const float*)d_in[8];
  float* out = (float*)d_out;

  // workspace carve-up (~68 MB)
  float* x      = (float*)d_ws;                 // N*D
  float* segout = x + (size_t)N_ * D_;          // N*D
  float* ls     = segout + (size_t)N_ * D_;     // N*H
  float* ld     = ls + (size_t)N_ * H_;         // N*H
  float* m      = ld + (size_t)N_ * H_;         // N*H
  float* ssum   = m + (size_t)N_ * H_;          // N*H

  for (int l = 0; l < NL_; ++l) {
    const float* hsrc = (l == 0) ? route_emb : out;

    gat_gemm_wmma<<<N_ / 128, 256, 0, stream>>>(
        hsrc, W + (size_t)l * D_ * D_, x);

    gat_logits<<<(N_ * H_ + 255) / 256, 256, 0, stream>>>(
        x, att_src + l * H_ * 32, att_dst + l * H_ * 32, ls, ld);

    gat_init<<<(N_ * D_ + 255) / 256, 256, 0, stream>>>(segout, m, ssum);

    gat_edge_max<<<(E_ + 255) / 256, 256, 0, stream>>>(adj, ls, ld, m);

    gat_edge_sum<<<(E_ + 255) / 256, 256, 0, stream>>>(adj, ls, ld, m, ssum);

    gat_edge_scatter<<<E_ / 8, 256, 0, stream>>>(adj, ls, ld, m, ssum, x, segout);

    gat_ln<<<N_ / 8, 256, 0, stream>>>(
        hsrc, segout, bias + l * D_, gamma + l * D_, beta + l * D_, out);
  }
}